// ChildSumTreeLSTM_53274774340141
// MI455X (gfx1250) — compile-verified
//
#include <hip/hip_runtime.h>

// Child-Sum Tree-LSTM for MI455X (gfx1250).
//  k_init      : zero hs/fcs accumulators + fx dummy row (replay-safe)
//  k_pre_gemm  : fused embed-gather + [W_ioux;W_fx] projection, V_WMMA_F32_16X16X4_F32
//  k_tree_scan : sequential scan, 1 workgroup x 1024 threads; W_fh cached in LDS
//                (257-float padded rows -> conflict-free), float4 W_iouh GEMV
//  k_out_gemm  : h_all @ W_out^T + b_out via WMMA f32 (store-BW bound: 512MB logits)
//  k_softmax   : per-row softmax over 32000, in place in d_out

#define N_NODES 4096
#define VOCAB   32000
#define IN_DIM  256
#define MEM_DIM 256
#define IOU_DIM 768
#define WFH_STRIDE 257   // 256 + 1 pad -> LDS bank (row + k) % 64, conflict-free

typedef __attribute__((ext_vector_type(2))) float v2f;
typedef __attribute__((ext_vector_type(8))) float v8f;

__device__ __forceinline__ float fsigmoid(float x) {
    return 1.0f / (1.0f + __expf(-x));
}
__device__ __forceinline__ float ftanh(float x) {
    x = fminf(fmaxf(x, -15.0f), 15.0f);
    float e = __expf(-2.0f * x);
    return (1.0f - e) / (1.0f + e);
}

// ---------------------------------------------------------------- init state
__global__ void k_init(float* hs, float* fcs, float* fx_dummy_row) {
    long i = (long)blockIdx.x * blockDim.x + threadIdx.x;
    long total = (long)(N_NODES + 1) * MEM_DIM;
    if (i < total) { hs[i] = 0.0f; fcs[i] = 0.0f; }
    if (i < MEM_DIM) fx_dummy_row[i] = 0.0f;
}

// ------------------------------------------- fused embed + input projections
// one wave per 16x16 tile of the [4096 x 1024] output (cols 0..767 -> ioux,
// cols 768..1023 -> fx). K = IN_DIM = 256 -> 64 wmma_f32_16x16x4 per tile.
__global__ __launch_bounds__(256)
void k_pre_gemm(const int* __restrict__ tokens, const float* __restrict__ emb,
                const float* __restrict__ W_ioux, const float* __restrict__ b_ioux,
                const float* __restrict__ W_fx,
                float* __restrict__ ioux_all, float* __restrict__ fx_all) {
    const int ntn = (IOU_DIM + MEM_DIM) / 16;              // 64 col tiles
    int wave = __builtin_amdgcn_readfirstlane(
        (int)((blockIdx.x * blockDim.x + threadIdx.x) >> 5));
    int tm = wave / ntn;                                    // scalar tile coords
    int tn = wave % ntn;
    if (tm >= N_NODES / 16) return;

    int lane = threadIdx.x & 31;
    int half = lane >> 4;                                   // 0: K{0,1}, 1: K{2,3}
    int l16  = lane & 15;
    int koff = half * 2;

    int m   = tm * 16 + l16;                                // A row for this lane
    int n0  = tn * 16;                                      // scalar
    int tok = tokens[m];
    const float* arow  = emb + (long)tok * IN_DIM;
    const float* wbase = (n0 < IOU_DIM)                     // scalar select
        ? (W_ioux + (long)n0 * IN_DIM)
        : (W_fx   + (long)(n0 - IOU_DIM) * IN_DIM);
    const float* wrow  = wbase + (long)l16 * IN_DIM;

    v8f acc = {};
#pragma unroll 8
    for (int k = 0; k < IN_DIM; k += 4) {
        v2f a = *(const v2f*)(arow + k + koff);
        v2f b = *(const v2f*)(wrow + k + koff);
        acc = __builtin_amdgcn_wmma_f32_16x16x4_f32(false, a, false, b,
                                                    (short)0, acc, false, false);
    }

    int col = n0 + l16;
    if (n0 < IOU_DIM) {                                     // scalar branch
        float bias = b_ioux[col];
#pragma unroll
        for (int v = 0; v < 8; ++v) {
            int row = tm * 16 + v + half * 8;
            ioux_all[(long)row * IOU_DIM + col] = acc[v] + bias;
        }
    } else {
        int cf = col - IOU_DIM;
#pragma unroll
        for (int v = 0; v < 8; ++v) {
            int row = tm * 16 + v + half * 8;
            fx_all[(long)row * MEM_DIM + cf] = acc[v];
        }
    }
}

// ------------------------------------------------------- sequential tree scan
// Single workgroup (32 waves on one WGP). Component t of hs/fcs is owned
// exclusively by thread t -> plain RMW scatter-adds, no atomics needed.
// Dynamic LDS: W_fh (256x257 padded, 257KB) + step buffers (~5KB) < 320KB/WGP.
__global__ __launch_bounds__(1024)
void k_tree_scan(const float* __restrict__ ioux_all, const float* __restrict__ fx_all,
                 const int* __restrict__ parents,
                 const float* __restrict__ W_iouh, const float* __restrict__ b_iouh,
                 const float* __restrict__ W_fh,
                 const float* __restrict__ b_fx, const float* __restrict__ b_fh,
                 float* __restrict__ hs, float* __restrict__ fcs,
                 float* __restrict__ c_out, float* __restrict__ h_out) {
    extern __shared__ float smem[];
    float* s_wfh = smem;                                    // 256 * 257
    float* s_h   = smem + MEM_DIM * WFH_STRIDE;             // 256
    float* s_c   = s_h + MEM_DIM;                           // 256
    float* s_iou = s_c + MEM_DIM;                           // 768

    int t = threadIdx.x;
    float bi = (t < IOU_DIM) ? b_iouh[t] : 0.0f;
    float fb = (t < MEM_DIM) ? (b_fx[t] + b_fh[t]) : 0.0f;

    // one-time: stage W_fh into padded LDS
    for (int i = t; i < MEM_DIM * MEM_DIM; i += 1024) {
        int r = i >> 8, c = i & 255;
        s_wfh[r * WFH_STRIDE + c] = W_fh[i];
    }
    __syncthreads();

    for (int idx = 0; idx < N_NODES; ++idx) {
        // phase 1: broadcast child-sum h into LDS
        if (t < MEM_DIM) s_h[t] = hs[(long)idx * MEM_DIM + t];
        __syncthreads();

        // phase 2: iou = ioux_all[idx] + W_iouh @ h + b_iouh   (768 rows, f4 loads)
        if (t < IOU_DIM) {
            const float4* wr = (const float4*)(W_iouh + (long)t * MEM_DIM);
            const float4* hv = (const float4*)s_h;
            float acc = bi + ioux_all[(long)idx * IOU_DIM + t];
#pragma unroll 8
            for (int k = 0; k < MEM_DIM / 4; ++k) {
                float4 w  = wr[k];
                float4 h4 = hv[k];
                acc = fmaf(w.x, h4.x, acc);
                acc = fmaf(w.y, h4.y, acc);
                acc = fmaf(w.z, h4.z, acc);
                acc = fmaf(w.w, h4.w, acc);
            }
            s_iou[t] = acc;
        }
        __syncthreads();

        // phase 3: gates, c, h
        if (t < MEM_DIM) {
            float ig = fsigmoid(s_iou[t]);
            float og = fsigmoid(s_iou[MEM_DIM + t]);
            float ug = ftanh(s_iou[2 * MEM_DIM + t]);
            float c  = fmaf(ig, ug, fcs[(long)idx * MEM_DIM + t]);
            float h  = og * ftanh(c);
            s_c[t] = c;
            s_h[t] = h;                             // now holds current node h
            c_out[(long)idx * MEM_DIM + t] = c;
            h_out[(long)idx * MEM_DIM + t] = h;
        }
        __syncthreads();

        // phase 4+5: f = sigmoid(W_fh@h + fx[p] + b), scatter-add into parent.
        // s_wfh row stride 257 -> lane bank (t + k) % 64, conflict-free;
        // s_h[k] is an LDS broadcast.
        if (t < MEM_DIM) {
            const float* wr = s_wfh + t * WFH_STRIDE;
            float acc = 0.0f;
#pragma unroll 16
            for (int k = 0; k < MEM_DIM; ++k) acc = fmaf(wr[k], s_h[k], acc);
            int p = parents[idx];
            float f = fsigmoid(acc + fx_all[(long)p * MEM_DIM + t] + fb);
            hs[(long)p * MEM_DIM + t]  += s_h[t];
            fcs[(long)p * MEM_DIM + t] += f * s_c[t];
        }
        __syncthreads();
    }
}

// ------------------------------------------------- output projection (WMMA)
__global__ __launch_bounds__(256)
void k_out_gemm(const float* __restrict__ h_all, const float* __restrict__ W_out,
                const float* __restrict__ b_out, float* __restrict__ logits) {
    const int ntn = VOCAB / 16;                             // 2000 col tiles
    int wave = __builtin_amdgcn_readfirstlane(
        (int)((blockIdx.x * blockDim.x + threadIdx.x) >> 5));
    int tm = wave / ntn;
    int tn = wave % ntn;
    if (tm >= N_NODES / 16) return;

    int lane = threadIdx.x & 31;
    int half = lane >> 4;
    int l16  = lane & 15;
    int koff = half * 2;

    const float* arow = h_all + (long)(tm * 16 + l16) * MEM_DIM;
    const float* wrow = W_out + (long)(tn * 16) * MEM_DIM + (long)l16 * MEM_DIM;

    v8f acc = {};
#pragma unroll 8
    for (int k = 0; k < MEM_DIM; k += 4) {
        v2f a = *(const v2f*)(arow + k + koff);
        v2f b = *(const v2f*)(wrow + k + koff);
        acc = __builtin_amdgcn_wmma_f32_16x16x4_f32(false, a, false, b,
                                                    (short)0, acc, false, false);
    }

    int col = tn * 16 + l16;
    float bias = b_out[col];
#pragma unroll
    for (int v = 0; v < 8; ++v) {
        int row = tm * 16 + v + half * 8;
        logits[(long)row * VOCAB + col] = acc[v] + bias;
    }
}

// -------------------------------------------------------- row softmax (32000)
__global__ __launch_bounds__(512)
void k_softmax(float* __restrict__ soft) {
    __shared__ float red[512];
    int t = threadIdx.x;
    float* p = soft + (long)blockIdx.x * VOCAB;

    float m = -3.0e38f;
    for (int i = t; i < VOCAB; i += 512) m = fmaxf(m, p[i]);
    red[t] = m; __syncthreads();
    for (int s = 256; s > 0; s >>= 1) {
        if (t < s) red[t] = fmaxf(red[t], red[t + s]);
        __syncthreads();
    }
    m = red[0]; __syncthreads();

    float sum = 0.0f;
    for (int i = t; i < VOCAB; i += 512) sum += __expf(p[i] - m);
    red[t] = sum; __syncthreads();
    for (int s = 256; s > 0; s >>= 1) {
        if (t < s) red[t] += red[t + s];
        __syncthreads();
    }
    float inv = 1.0f / red[0];

    for (int i = t; i < VOCAB; i += 512) p[i] = __expf(p[i] - m) * inv;
}

// ---------------------------------------------------------------------------
extern "C" void kernel_launch(void* const* d_in, const int* in_sizes, int n_in,
                              void* d_out, int out_size, void* d_ws, size_t ws_size,
                              hipStream_t stream) {
    const int*   tokens  = (const int*)  d_in[0];
    const int*   parents = (const int*)  d_in[1];
    const float* emb     = (const float*)d_in[2];
    const float* W_ioux  = (const float*)d_in[3];
    const float* b_ioux  = (const float*)d_in[4];
    const float* W_iouh  = (const float*)d_in[5];
    const float* b_iouh  = (const float*)d_in[6];
    const float* W_fx    = (const float*)d_in[7];
    const float* b_fx    = (const float*)d_in[8];
    const float* W_fh    = (const float*)d_in[9];
    const float* b_fh    = (const float*)d_in[10];
    const float* W_out   = (const float*)d_in[11];
    const float* b_out   = (const float*)d_in[12];

    float* out   = (float*)d_out;
    float* c_all = out;                                   // [4096, 256]
    float* h_all = out + (long)N_NODES * MEM_DIM;         // [4096, 256]
    float* soft  = out + 2L * N_NODES * MEM_DIM;          // [4096, 32000]

    float* ws       = (float*)d_ws;
    float* ioux_all = ws;                                           // 4096*768
    float* fx_all   = ioux_all + (long)N_NODES * IOU_DIM;           // 4097*256 (padded)
    float* hs       = fx_all + (long)(N_NODES + 1) * MEM_DIM;       // 4097*256
    float* fcs      = hs + (long)(N_NODES + 1) * MEM_DIM;           // 4097*256

    {   // zero accumulators + dummy fx row (must re-run every launch)
        long total = (long)(N_NODES + 1) * MEM_DIM;
        int blocks = (int)((total + 255) / 256);
        k_init<<<blocks, 256, 0, stream>>>(hs, fcs, fx_all + (long)N_NODES * MEM_DIM);
    }
    {   // input projections: (4096/16)*(1024/16) = 16384 waves, 8 waves/block
        int waves = (N_NODES / 16) * ((IOU_DIM + MEM_DIM) / 16);
        k_pre_gemm<<<waves / 8, 256, 0, stream>>>(tokens, emb, W_ioux, b_ioux, W_fx,
                                                  ioux_all, fx_all);
    }
    {   // scan: dynamic LDS = W_fh padded (256*257) + h/c/iou buffers
        size_t sh = (size_t)(MEM_DIM * WFH_STRIDE + MEM_DIM + MEM_DIM + IOU_DIM)
                    * sizeof(float);                       // 268,288 B < 320KB
        k_tree_scan<<<1, 1024, sh, stream>>>(ioux_all, fx_all, parents,
                                             W_iouh, b_iouh, W_fh, b_fx, b_fh,
                                             hs, fcs, c_all, h_all);
    }
    {   // output GEMM: (4096/16)*(32000/16) = 512000 waves, 8 waves/block
        long waves = (long)(N_NODES / 16) * (VOCAB / 16);
        k_out_gemm<<<(int)(waves / 8), 256, 0, stream>>>(h_all, W_out, b_out, soft);
    }
    k_softmax<<<N_NODES, 512, 0, stream>>>(soft);
}